// VectorQuantizer_39797166964740
// MI455X (gfx1250) — compile-verified
//
#include <hip/hip_runtime.h>

typedef __attribute__((ext_vector_type(2))) float v2f;
typedef __attribute__((ext_vector_type(8))) float v8f;

#define NUM_EMB 1024
#define DIM 64
#define ROWS_PER_BLOCK 128
#define THREADS 256
#define CB_STRIDE 66   // even -> 8B-aligned ds_load_b64; bank advance 2/row -> conflict-free
#define Q_STRIDE 65    // odd -> conflict-free transpose reads
#define T_LEN 2048

constexpr int N_ROWS = 32 * T_LEN;          // 65536
constexpr int QOUT   = N_ROWS * DIM;        // 4194304  (quantized elements)
// out layout: [quantized (QOUT)] [loss (1)] [indices (N_ROWS)]

__global__ __launch_bounds__(THREADS)
void vq_main(const float* __restrict__ x, const float* __restrict__ cb,
             float* __restrict__ out, float* __restrict__ partial) {
    extern __shared__ float smem[];
    float* lds_cb = smem;                                 // 1024*66
    float* cnorm  = lds_cb + NUM_EMB * CB_STRIDE;         // 1024
    float* qstage = cnorm + NUM_EMB;                      // 128*65
    float* red    = qstage + ROWS_PER_BLOCK * Q_STRIDE;   // 256

    const int tid = threadIdx.x;

    // Stage codebook into LDS (coalesced global reads)
    for (int e = tid; e < NUM_EMB * DIM; e += THREADS) {
        int rw = e >> 6, cl = e & 63;
        lds_cb[rw * CB_STRIDE + cl] = cb[e];
    }
    __syncthreads();
    // Per-code squared norms
    for (int i = tid; i < NUM_EMB; i += THREADS) {
        float s = 0.f;
        #pragma unroll
        for (int d = 0; d < DIM; ++d) {
            float v = lds_cb[i * CB_STRIDE + d];
            s += v * v;
        }
        cnorm[i] = s;
    }
    __syncthreads();

    const int wave = tid >> 5;
    const int lane = tid & 31;
    const int m    = lane & 15;
    const int half = lane >> 4;
    const int row0 = blockIdx.x * ROWS_PER_BLOCK + wave * 16;

    // A fragments: 16x4 f32 tiles covering K=D=64.
    // ISA layout: lanes 0-15 -> M=lane, K={0,1}; lanes 16-31 -> M=lane-16, K={2,3}.
    v2f a[16];
    const float* xr = x + (size_t)(row0 + m) * DIM + 2 * half;
    #pragma unroll
    for (int kk = 0; kk < 16; ++kk) a[kk] = *(const v2f*)(xr + 4 * kk);

    float bests[8];
    int   bidx[8];
    #pragma unroll
    for (int r = 0; r < 8; ++r) { bests[r] = -3.4e38f; bidx[r] = 0; }

    // Loop over code tiles: score = 2*x.e - ||e||^2 (maximize == argmin distance)
    for (int n0 = 0; n0 < NUM_EMB; n0 += 16) {
        v8f acc = {};
        const float* bl = lds_cb + (n0 + m) * CB_STRIDE + 2 * half; // B = codebook^T, mirrored A layout
        #pragma unroll
        for (int kk = 0; kk < 16; ++kk) {
            v2f b = *(const v2f*)(bl + 4 * kk);
            acc = __builtin_amdgcn_wmma_f32_16x16x4_f32(
                false, a[kk], false, b, (short)0, acc, false, false);
        }
        const float cn = cnorm[n0 + m];
        const int   n  = n0 + m;
        #pragma unroll
        for (int r = 0; r < 8; ++r) {
            float s = 2.f * acc[r] - cn;
            bool better = (s > bests[r]) || (s == bests[r] && n < bidx[r]);
            bests[r] = better ? s : bests[r];
            bidx[r]  = better ? n : bidx[r];
        }
    }

    // Cross-lane argmax within each 16-lane half (rows r+8*half)
    #pragma unroll
    for (int r = 0; r < 8; ++r) {
        float s = bests[r];
        int   i = bidx[r];
        #pragma unroll
        for (int off = 8; off; off >>= 1) {
            float os = __shfl_xor(s, off, 16);
            int   oi = __shfl_xor(i, off, 16);
            bool better = (os > s) || (os == s && oi < i);
            s = better ? os : s;
            i = better ? oi : i;
        }
        bests[r] = s;
        bidx[r]  = i;
    }

    // Write indices, stage quantized rows to LDS, accumulate loss partial
    float lsum = 0.f;
    float* out_idx = out + QOUT + 1;
    #pragma unroll
    for (int r = 0; r < 8; ++r) {
        const int l = r + 8 * half;          // local row in wave tile
        const int g = row0 + l;              // global row
        const int q = bidx[r];               // winning code (same in all 16 lanes of half)
        if (m == 0) out_idx[g] = (float)q;
        const int lrow = wave * 16 + l;      // row within block stage
        #pragma unroll
        for (int j = 0; j < 4; ++j) {
            float qv = lds_cb[q * CB_STRIDE + 4 * m + j];
            qstage[lrow * Q_STRIDE + 4 * m + j] = qv;
            float xv = x[(size_t)g * DIM + 4 * m + j];
            float df = qv - xv;
            lsum += df * df;
        }
    }

    // Block-wide loss reduction (deterministic, no atomics)
    red[tid] = lsum;
    __syncthreads();
    for (int s2 = THREADS / 2; s2 > 0; s2 >>= 1) {
        if (tid < s2) red[tid] += red[tid + s2];
        __syncthreads();
    }
    if (tid == 0) partial[blockIdx.x] = red[0];

    // Coalesced transposed store: out[b, d, t] — 128 consecutive t per d
    const int base = blockIdx.x * ROWS_PER_BLOCK;
    const int t0   = base % T_LEN;
    const int bImg = base / T_LEN;
    float* qo = out + (size_t)bImg * DIM * T_LEN;
    for (int e = tid; e < DIM * ROWS_PER_BLOCK; e += THREADS) {
        int d  = e >> 7;        // /128
        int tl = e & 127;
        qo[d * T_LEN + t0 + tl] = qstage[tl * Q_STRIDE + d];
    }
}

__global__ __launch_bounds__(256)
void vq_loss(const float* __restrict__ partial, float* __restrict__ out) {
    __shared__ float red[256];
    float s = partial[threadIdx.x] + partial[threadIdx.x + 256];
    red[threadIdx.x] = s;
    __syncthreads();
    for (int k = 128; k > 0; k >>= 1) {
        if (threadIdx.x < k) red[threadIdx.x] += red[threadIdx.x + k];
        __syncthreads();
    }
    if (threadIdx.x == 0)
        out[QOUT] = 1.25f * red[0] / (float)(N_ROWS * DIM);  // (1 + 0.25) * mean((q-x)^2)
}

extern "C" void kernel_launch(void* const* d_in, const int* in_sizes, int n_in,
                              void* d_out, int out_size, void* d_ws, size_t ws_size,
                              hipStream_t stream) {
    (void)in_sizes; (void)n_in; (void)out_size; (void)ws_size;
    const float* x  = (const float*)d_in[0];
    const float* cb = (const float*)d_in[1];
    float* out      = (float*)d_out;
    float* partial  = (float*)d_ws;   // 512 floats

    const size_t lds_bytes =
        (size_t)(NUM_EMB * CB_STRIDE + NUM_EMB + ROWS_PER_BLOCK * Q_STRIDE + THREADS) * sizeof(float);

    const int blocks = N_ROWS / ROWS_PER_BLOCK;  // 512
    vq_main<<<blocks, THREADS, lds_bytes, stream>>>(x, cb, out, partial);
    vq_loss<<<1, 256, 0, stream>>>(partial, out);
}